// VectorQuantizerEMA_multi_head_revival_56813827391975
// MI455X (gfx1250) — compile-verified
//
#include <hip/hip_runtime.h>
#include <hip/hip_bf16.h>

typedef float v2f __attribute__((ext_vector_type(2)));
typedef float v8f __attribute__((ext_vector_type(8)));

#define H_    4
#define HD    64
#define K_    2048
#define D_    256
#define NTOK  16384                    /* b*s = 8*2048 */
#define QSIZE (NTOK * D_)              /* 4194304 */
#define DIFF_OFF QSIZE
#define IDX_OFF  (QSIZE + 1)
#define KC    128                      /* codes per LDS chunk */
#define NCHUNK (K_ / KC)               /* 16 */
#define TILES_PER_CHUNK (KC / 16)      /* 8 */
#define WAVES_PER_BLOCK 8
#define TOK_PER_WAVE  32               /* two 16-row A tiles per wave */
#define TOK_PER_BLOCK (WAVES_PER_BLOCK * TOK_PER_WAVE) /* 256 */
#define NWAVES_TOTAL (H_ * (NTOK / TOK_PER_WAVE))      /* 2048 */

// ---------------------------------------------------------------------------
// Kernel 1: precompute ||E_k||^2 per head/code into workspace (H*K floats).
// ---------------------------------------------------------------------------
__global__ void vq_ee_kernel(const float* __restrict__ e, float* __restrict__ ee) {
    int i = blockIdx.x * blockDim.x + threadIdx.x;   // 0 .. H*K-1
    int h = i / K_;
    int k = i - h * K_;
    const float* p = e + (size_t)h * HD * K_ + k;
    float s = 0.0f;
#pragma unroll 8
    for (int d = 0; d < HD; ++d) {
        float v = p[(size_t)d * K_];
        s += v * v;
    }
    ee[i] = s;
}

// ---------------------------------------------------------------------------
// Kernel 2: main VQ kernel.  Block = 256 threads = 8 waves, one head,
// 256 tokens (32 per wave as two 16-row A tiles).  E chunks staged in LDS;
// each B load feeds two independent fp32 WMMA chains (ILP + half the LDS
// traffic per FLOP); per-lane running argmin; gather + quantize + |x-q|.
// ---------------------------------------------------------------------------
__launch_bounds__(256)
__global__ void vq_main_kernel(const float* __restrict__ x,
                               const float* __restrict__ e,
                               const float* __restrict__ ee,
                               float* __restrict__ out,
                               float* __restrict__ partial) {
    __shared__ float ldsB[HD * KC];                       // 32 KB E chunk
    __shared__ int   sidx[WAVES_PER_BLOCK][TOK_PER_WAVE]; // 1 KB winning idx

    const int tid  = threadIdx.x;
    const int wave = tid >> 5;
    const int lane = tid & 31;
    const int half = lane >> 4;       // WMMA half: lanes 0-15 vs 16-31
    const int lid  = lane & 15;

    const int h    = blockIdx.x / (NTOK / TOK_PER_BLOCK);
    const int tb   = blockIdx.x % (NTOK / TOK_PER_BLOCK);
    const int tok0 = tb * TOK_PER_BLOCK + wave * TOK_PER_WAVE;

    // ---- Load two A tiles (tokens tok0..+15 and tok0+16..+31).
    // ISA layout for 32-bit A 16x4: lane = M (row), K = 2*half + vgpr.
    v2f a0[16], a1[16];
    {
        const float* xr0 = x + (size_t)(tok0 + lid) * D_ + h * HD + 2 * half;
        const float* xr1 = xr0 + 16 * D_;
#pragma unroll
        for (int j = 0; j < 16; ++j) {
            a0[j] = *(const v2f*)(xr0 + j * 4);
            a1[j] = *(const v2f*)(xr1 + j * 4);
        }
    }

    float minv0[8], minv1[8];
    int   mini0[8], mini1[8];
#pragma unroll
    for (int r = 0; r < 8; ++r) {
        minv0[r] = 3.0e38f; mini0[r] = 0;
        minv1[r] = 3.0e38f; mini1[r] = 0;
    }

    for (int cc = 0; cc < NCHUNK; ++cc) {
        const int cbase = cc * KC;
        __syncthreads();
        // ---- Stage E[h][0..63][cbase..cbase+127] into LDS (float4 loads).
        {
            const float4* src = (const float4*)(e + (size_t)h * HD * K_ + cbase);
#pragma unroll
            for (int i = 0; i < 8; ++i) {
                int f4 = tid * 8 + i;        // 0..2047 float4s
                int k  = f4 >> 5;            // row (32 float4 per row)
                int c4 = f4 & 31;
                float4 v = src[(size_t)k * (K_ / 4) + c4];
                *(float4*)&ldsB[k * KC + c4 * 4] = v;
            }
        }
        __syncthreads();
        if (cc + 1 < NCHUNK)   // hint next chunk toward cache
            __builtin_prefetch(e + (size_t)h * HD * K_ + cbase + KC + tid * 16, 0, 0);

#pragma unroll 1
        for (int ct = 0; ct < TILES_PER_CHUNK; ++ct) {
            const int c0    = cbase + ct * 16;
            const float eec = ee[h * K_ + c0 + lid];

            v8f acc0 = {0.f, 0.f, 0.f, 0.f, 0.f, 0.f, 0.f, 0.f};
            v8f acc1 = {0.f, 0.f, 0.f, 0.f, 0.f, 0.f, 0.f, 0.f};
#pragma unroll
            for (int j = 0; j < 16; ++j) {
                // B 4x16 layout mirrors A: lane = N (col), K = 2*half + vgpr.
                const int krow = j * 4 + 2 * half;
                v2f bb;
                bb.x = ldsB[krow * KC + ct * 16 + lid];
                bb.y = ldsB[(krow + 1) * KC + ct * 16 + lid];
                acc0 = __builtin_amdgcn_wmma_f32_16x16x4_f32(
                    false, a0[j], false, bb, (short)0, acc0, false, false);
                acc1 = __builtin_amdgcn_wmma_f32_16x16x4_f32(
                    false, a1[j], false, bb, (short)0, acc1, false, false);
            }
            // D layout: VGPR r -> row (r + 8*half), lane -> column lid.
#pragma unroll
            for (int r = 0; r < 8; ++r) {
                float d0 = eec - 2.0f * acc0[r];
                float d1 = eec - 2.0f * acc1[r];
                if (d0 < minv0[r]) { minv0[r] = d0; mini0[r] = c0 + lid; }
                if (d1 < minv1[r]) { minv1[r] = d1; mini1[r] = c0 + lid; }
            }
        }
    }

    // ---- Cross-lane argmin within each 16-lane half (first-index tiebreak).
    // Lanes 0/16 then publish each half's token results to LDS scratch.
#pragma unroll
    for (int r = 0; r < 8; ++r) {
        float v0 = minv0[r]; int i0 = mini0[r];
        float v1 = minv1[r]; int i1 = mini1[r];
#pragma unroll
        for (int m = 8; m >= 1; m >>= 1) {
            float ov0 = __shfl_xor(v0, m, 32);
            int   oi0 = __shfl_xor(i0, m, 32);
            float ov1 = __shfl_xor(v1, m, 32);
            int   oi1 = __shfl_xor(i1, m, 32);
            if (ov0 < v0 || (ov0 == v0 && oi0 < i0)) { v0 = ov0; i0 = oi0; }
            if (ov1 < v1 || (ov1 == v1 && oi1 < i1)) { v1 = ov1; i1 = oi1; }
        }
        if (lid == 0) {                       // lanes 0 and 16
            sidx[wave][half * 8 + r]      = i0;   // tile 0: tokens 0..15
            sidx[wave][16 + half * 8 + r] = i1;   // tile 1: tokens 16..31
        }
    }
    // (compiler inserts the wave-local s_wait_dscnt for the LDS RAW below)

    // ---- Write indices ([b,s,H] layout, stored as float).
    {
        int n = tok0 + lane;
        out[IDX_OFF + (size_t)n * H_ + h] = (float)sidx[wave][lane];
    }

    // ---- Gather codes -> quantize output; accumulate |x - q|.
    float s = 0.0f;
#pragma unroll 1
    for (int t = 0; t < TOK_PER_WAVE; ++t) {
        const int n  = tok0 + t;
        const int ix = sidx[wave][t];
#pragma unroll
        for (int rep = 0; rep < 2; ++rep) {
            const int d = lane + rep * 32;
            const float qv = e[(size_t)(h * HD + d) * K_ + ix];
            const float xv = x[(size_t)n * D_ + h * HD + d];
            out[(size_t)n * D_ + h * HD + d] = qv;
            s += fabsf(xv - qv);
        }
    }
#pragma unroll
    for (int m = 16; m >= 1; m >>= 1) s += __shfl_xor(s, m, 32);
    if (lane == 0)
        partial[blockIdx.x * WAVES_PER_BLOCK + wave] = s;
}

// ---------------------------------------------------------------------------
// Kernel 3: deterministic fixed-order reduction of 2048 partials -> diff.
// ---------------------------------------------------------------------------
__global__ void vq_reduce_kernel(const float* __restrict__ partial,
                                 float* __restrict__ out) {
    __shared__ float sm[256];
    float s = 0.0f;
#pragma unroll 1
    for (int i = 0; i < NWAVES_TOTAL / 256; ++i)
        s += partial[threadIdx.x * (NWAVES_TOTAL / 256) + i];
    sm[threadIdx.x] = s;
    __syncthreads();
    for (int m = 128; m >= 1; m >>= 1) {
        if ((int)threadIdx.x < m) sm[threadIdx.x] += sm[threadIdx.x + m];
        __syncthreads();
    }
    if (threadIdx.x == 0)
        out[DIFF_OFF] = sm[0] / (float)QSIZE;
}

// ---------------------------------------------------------------------------
extern "C" void kernel_launch(void* const* d_in, const int* in_sizes, int n_in,
                              void* d_out, int out_size, void* d_ws, size_t ws_size,
                              hipStream_t stream) {
    const float* x = (const float*)d_in[0];   // [8,2048,256] fp32
    const float* e = (const float*)d_in[1];   // [4,64,2048] fp32
    float* out     = (float*)d_out;
    float* ee      = (float*)d_ws;            // H*K = 8192 floats
    float* partial = ee + H_ * K_;            // 2048 floats

    vq_ee_kernel<<<(H_ * K_) / 256, 256, 0, stream>>>(e, ee);
    vq_main_kernel<<<H_ * (NTOK / TOK_PER_BLOCK), 256, 0, stream>>>(x, e, ee, out, partial);
    vq_reduce_kernel<<<1, 256, 0, stream>>>(partial, out);
}